// GraphLED_61649960567367
// MI455X (gfx1250) — compile-verified
//
#include <hip/hip_runtime.h>
#include <hip/hip_bf16.h>
#include <cstdint>

// ---------------- model constants (match reference) ----------------
#define BB    2
#define TT    4
#define HIDD  64
#define LATD  32
#define NHEAD 4
#define DHD   8
#define EPSLN 1e-5f

typedef __attribute__((ext_vector_type(2))) float v2f;
typedef __attribute__((ext_vector_type(8))) float v8f;

// ---------------- wave32 helpers ----------------
__device__ __forceinline__ float wave_sum32(float v) {
  v += __shfl_xor(v, 16, 32);
  v += __shfl_xor(v, 8, 32);
  v += __shfl_xor(v, 4, 32);
  v += __shfl_xor(v, 2, 32);
  v += __shfl_xor(v, 1, 32);
  return v;
}

__device__ __forceinline__ void atomic_fadd(float* p, float v) {
  // HIP intrinsic -> global_atomic_add_f32 on gfx1250 (no CAS loop)
  unsafeAtomicAdd(p, v);
}

// ---------------- degree (includes self loop) ----------------
__global__ void deg_init_kernel(float* __restrict__ deg, int N) {
  int i = blockIdx.x * blockDim.x + threadIdx.x;
  if (i < N) deg[i] = 1.0f;  // self loop contributes 1
}

__global__ void deg_scatter_kernel(const int* __restrict__ dst, float* __restrict__ deg, int E) {
  int e = blockIdx.x * blockDim.x + threadIdx.x;
  if (e < E) atomic_fadd(&deg[dst[e]], 1.0f);
}

// ---------------- encoder input projection: [rows,3] @ [3,64] + b ----------------
__global__ void enc_in_kernel(const float* __restrict__ X, const float* __restrict__ W,
                              const float* __restrict__ Bv, float* __restrict__ Y, int rows) {
  int idx = blockIdx.x * blockDim.x + threadIdx.x;
  if (idx >= rows * HIDD) return;
  int r = idx >> 6, f = idx & 63;
  const float* x = X + (size_t)r * 3;
  Y[idx] = fmaf(x[0], W[f], fmaf(x[1], W[64 + f], fmaf(x[2], W[128 + f], Bv[f])));
}

// ---------------- GNN: agg = x + relu(W_e*1 + b_e)  (self loop, ea==1), float4 ----------------
__global__ void agg_init_kernel(const float* __restrict__ x, const float* __restrict__ eW,
                                const float* __restrict__ eB, float* __restrict__ agg, int total4) {
  int i4 = blockIdx.x * blockDim.x + threadIdx.x;
  if (i4 >= total4) return;
  int idx = i4 * 4;
  int f = idx & 63;
  float4 w  = *(const float4*)(eW + f);
  float4 b  = *(const float4*)(eB + f);
  float4 xv = *(const float4*)(x + idx);
  float4 o;
  o.x = xv.x + fmaxf(w.x + b.x, 0.f);
  o.y = xv.y + fmaxf(w.y + b.y, 0.f);
  o.z = xv.z + fmaxf(w.z + b.z, 0.f);
  o.w = xv.w + fmaxf(w.w + b.w, 0.f);
  *(float4*)(agg + idx) = o;
}

// ---------------- GNN: per-edge scatter, shared across G graphs ----------------
// one block (64 threads = 64 features) per edge
__global__ void edge_scatter_kernel(const int* __restrict__ src, const int* __restrict__ dst,
                                    const float* __restrict__ ea, const float* __restrict__ eW,
                                    const float* __restrict__ eB, const float* __restrict__ x,
                                    float* __restrict__ agg, int N, int G) {
  int e = blockIdx.x;
  int f = threadIdx.x;
  int s = src[e], d = dst[e];
  float a = ea[e];
  float ef = fmaxf(fmaf(a, eW[f], eB[f]), 0.f);  // relu(ea*W0 + b), shared across graphs
  size_t so   = (size_t)s * HIDD + f;
  size_t dofs = (size_t)d * HIDD + f;
  size_t stride = (size_t)N * HIDD;
  for (int g = 0; g < G; ++g)
    atomic_fadd(&agg[dofs + (size_t)g * stride], x[so + (size_t)g * stride] + ef);
}

// ---------------- GNN node update: (agg@W)/deg + b -> LN -> relu ----------------
// 128 threads = 4 waves; each wave one 16x16 tile via 16 chained V_WMMA_F32_16X16X4_F32.
// Weights staged into LDS via global_load_async_to_lds_b128 (ASYNCcnt path);
// 1/deg row scaling commuted to after the matmul (4 divides/wave, not 1024/tile).
__global__ void gnn_node_kernel(const float* __restrict__ agg, const float* __restrict__ deg,
                                const float* __restrict__ Wn, const float* __restrict__ bn,
                                const float* __restrict__ lng, const float* __restrict__ lnb,
                                float* __restrict__ xout, int N, int ntiles) {
  __shared__ __align__(16) float xs[16 * 68];   // A tile  [16][64] padded
  __shared__ __align__(16) float ws[64 * 64];   // weights [64][64] (unpadded: B reads conflict-free)
  __shared__ __align__(16) float hs[16 * 68];   // result  [16][64] padded
  int g  = blockIdx.x / ntiles;
  int n0 = (blockIdx.x % ntiles) * 16;
  int tid = threadIdx.x;
  size_t base = ((size_t)g * N + n0) * HIDD;

  // async-stage the 64x64 weight matrix into LDS: 128 threads x 8 x b128
  {
    const float* wp = Wn + tid * 4;
    uint32_t lo = (uint32_t)(uintptr_t)(&ws[tid * 4]);
#pragma unroll
    for (int i = 0; i < 8; ++i) {
      asm volatile("global_load_async_to_lds_b128 %0, %1, off"
                   :: "v"(lo), "v"(wp) : "memory");
      wp += 512;          // 128 threads * 4 floats
      lo += 512 * 4;      // bytes
    }
  }
  // A tile is contiguous (rows stride HIDD): float4 copy into padded LDS rows
#pragma unroll
  for (int i = tid * 4; i < 16 * 64; i += 128 * 4) {
    int r = i >> 6, c = i & 63;
    *(float4*)(xs + r * 68 + c) = *(const float4*)(agg + base + i);
  }
  asm volatile("s_wait_asynccnt 0x0" ::: "memory");
  __syncthreads();

  int wid = tid >> 5, lane = tid & 31;
  int mn  = lane & 15;
  int off = (lane < 16) ? 0 : 2;     // f32 16x16x4 A/B lane->K mapping
  v8f acc = {};
#pragma unroll
  for (int kk = 0; kk < 16; ++kk) {
    int k0 = kk * 4 + off;
    v2f a, b;
    a.x = xs[mn * 68 + k0];              a.y = xs[mn * 68 + k0 + 1];
    b.x = ws[k0 * 64 + wid * 16 + mn];   b.y = ws[(k0 + 1) * 64 + wid * 16 + mn];
    acc = __builtin_amdgcn_wmma_f32_16x16x4_f32(false, a, false, b, (short)0, acc,
                                                false, false);
  }
  int rbase = (lane < 16) ? 0 : 8;
#pragma unroll
  for (int r = 0; r < 8; ++r)
    hs[(rbase + r) * 68 + wid * 16 + mn] = acc[r];
  __syncthreads();

  // scatter-mean scale + bias + LayerNorm(64) + relu; 4 rows per wave
#pragma unroll
  for (int rr = 0; rr < 4; ++rr) {
    int r = wid * 4 + rr;
    float rd = 1.0f / deg[n0 + r];
    float v0 = hs[r * 68 + lane]      * rd + bn[lane];
    float v1 = hs[r * 68 + lane + 32] * rd + bn[lane + 32];
    float s  = wave_sum32(v0 + v1);
    float sq = wave_sum32(v0 * v0 + v1 * v1);
    float mu  = s * (1.f / 64.f);
    float var = sq * (1.f / 64.f) - mu * mu;
    float rstd = rsqrtf(var + EPSLN);
    float o0 = fmaxf((v0 - mu) * rstd * lng[lane]      + lnb[lane], 0.f);
    float o1 = fmaxf((v1 - mu) * rstd * lng[lane + 32] + lnb[lane + 32], 0.f);
    xout[base + (size_t)r * HIDD + lane]      = o0;
    xout[base + (size_t)r * HIDD + lane + 32] = o1;
  }
}

// ---------------- generic WMMA linear: Y[M,NOUT] = act(X[M,K] @ W[K,NOUT] + b) ----------------
template <int K, int NOUT, int NT, bool RELU>
__global__ void wmma_linear_kernel(const float* __restrict__ X, int ldx,
                                   const float* __restrict__ W, const float* __restrict__ Bv,
                                   float* __restrict__ Y, int ldy, int M) {
  __shared__ __align__(16) float xs[16 * (K + 4)];
  int m0 = blockIdx.x * 16;
  int tid = threadIdx.x;
  for (int i = tid * 4; i < 16 * K; i += 32 * NT * 4) {
    int r = i / K, c = i % K;
    int row = m0 + r;
    float4 v = make_float4(0.f, 0.f, 0.f, 0.f);
    if (row < M) v = *(const float4*)(X + (size_t)row * ldx + c);
    *(float4*)(xs + r * (K + 4) + c) = v;
  }
  __syncthreads();

  int wid = tid >> 5, lane = tid & 31;
  int mn  = lane & 15;
  int off = (lane < 16) ? 0 : 2;
  for (int ct = wid; ct < NOUT / 16; ct += NT) {
    v8f acc = {};
#pragma unroll
    for (int kk = 0; kk < K / 4; ++kk) {
      int k0 = kk * 4 + off;
      v2f a, b;
      a.x = xs[mn * (K + 4) + k0];          a.y = xs[mn * (K + 4) + k0 + 1];
      b.x = W[k0 * NOUT + ct * 16 + mn];    b.y = W[(k0 + 1) * NOUT + ct * 16 + mn];
      acc = __builtin_amdgcn_wmma_f32_16x16x4_f32(false, a, false, b, (short)0, acc,
                                                  false, false);
    }
    int col = ct * 16 + mn;
    float bv = Bv[col];
    int rbase = (lane < 16) ? 0 : 8;
#pragma unroll
    for (int r = 0; r < 8; ++r) {
      int row = m0 + rbase + r;
      float v = acc[r] + bv;
      if (RELU) v = fmaxf(v, 0.f);
      if (row < M) Y[(size_t)row * ldy + col] = v;
    }
  }
}

// ---------------- fused q/k/v projections (one X-tile load, 6 waves) ----------------
__global__ void qkv_kernel(const float* __restrict__ X,
                           const float* __restrict__ Wq, const float* __restrict__ bq,
                           const float* __restrict__ Wk, const float* __restrict__ bk,
                           const float* __restrict__ Wv, const float* __restrict__ bv,
                           float* __restrict__ Q, float* __restrict__ Ko,
                           float* __restrict__ V) {
  __shared__ __align__(16) float xs[16 * 36];
  int m0 = blockIdx.x * 16;
  int tid = threadIdx.x;  // 192 = 6 waves
  for (int i = tid * 4; i < 16 * 32; i += 192 * 4) {
    int r = i >> 5, c = i & 31;
    *(float4*)(xs + r * 36 + c) = *(const float4*)(X + (size_t)(m0 + r) * LATD + c);
  }
  __syncthreads();

  int wid = tid >> 5, lane = tid & 31;
  int which = wid >> 1, ct = wid & 1;
  const float* W  = (which == 0) ? Wq : (which == 1) ? Wk : Wv;
  const float* Bb = (which == 0) ? bq : (which == 1) ? bk : bv;
  float*       Y  = (which == 0) ? Q  : (which == 1) ? Ko : V;
  int mn  = lane & 15;
  int off = (lane < 16) ? 0 : 2;
  v8f acc = {};
#pragma unroll
  for (int kk = 0; kk < 8; ++kk) {
    int k0 = kk * 4 + off;
    v2f a, b;
    a.x = xs[mn * 36 + k0];              a.y = xs[mn * 36 + k0 + 1];
    b.x = W[k0 * 32 + ct * 16 + mn];     b.y = W[(k0 + 1) * 32 + ct * 16 + mn];
    acc = __builtin_amdgcn_wmma_f32_16x16x4_f32(false, a, false, b, (short)0, acc,
                                                false, false);
  }
  int col = ct * 16 + mn;
  float bias = Bb[col];
  int rbase = (lane < 16) ? 0 : 8;
#pragma unroll
  for (int r = 0; r < 8; ++r)
    Y[(size_t)(m0 + rbase + r) * LATD + col] = acc[r] + bias;
}

// ---------------- [B*T, N, LAT] -> [B, N, T, LAT], float4 ----------------
__global__ void reorder_kernel(const float* __restrict__ Z, float* __restrict__ D, int N) {
  int i4 = blockIdx.x * blockDim.x + threadIdx.x;
  int total4 = BB * N * TT * (LATD / 4);
  if (i4 >= total4) return;
  int f  = (i4 & 7) * 4;
  int r  = i4 >> 3;             // (b*N+n)*T + t
  int t  = r & (TT - 1);
  int r2 = r >> 2;              // b*N + n
  int n  = r2 % N;
  int b  = r2 / N;
  *(float4*)(D + (size_t)r * LATD + f) =
      *(const float4*)(Z + (((size_t)(b * TT + t)) * N + n) * LATD + f);
}

// ---------------- tiny attention: T=4, DH=8, one thread per (seq, head) ----------------
__global__ void attn_kernel(const float* __restrict__ Q, const float* __restrict__ Kb,
                            const float* __restrict__ V, float* __restrict__ O, int M) {
  int idx = blockIdx.x * blockDim.x + threadIdx.x;
  if (idx >= M * NHEAD) return;
  int m = idx >> 2, hh = idx & 3;
  size_t base = (size_t)m * TT * LATD + hh * DHD;
  float q[TT][DHD], k[TT][DHD], v[TT][DHD];
#pragma unroll
  for (int t = 0; t < TT; ++t)
#pragma unroll
    for (int j = 0; j < DHD; ++j) {
      q[t][j] = Q[base + t * LATD + j];
      k[t][j] = Kb[base + t * LATD + j];
      v[t][j] = V[base + t * LATD + j];
    }
  const float scale = 0.35355339059327373f;  // 1/sqrt(DH)
#pragma unroll
  for (int t = 0; t < TT; ++t) {
    float s[TT], mx = -1e30f;
#pragma unroll
    for (int u = 0; u < TT; ++u) {
      float d = 0.f;
#pragma unroll
      for (int j = 0; j < DHD; ++j) d = fmaf(q[t][j], k[u][j], d);
      s[u] = d * scale;
      mx = fmaxf(mx, s[u]);
    }
    float den = 0.f;
#pragma unroll
    for (int u = 0; u < TT; ++u) { s[u] = __expf(s[u] - mx); den += s[u]; }
    float rden = 1.f / den;
#pragma unroll
    for (int j = 0; j < DHD; ++j) {
      float o = 0.f;
#pragma unroll
      for (int u = 0; u < TT; ++u) o = fmaf(s[u], v[u][j], o);
      O[base + t * LATD + j] = o * rden;
    }
  }
}

// ---------------- post-norm residual LN, dim 32 == one wave per row ----------------
__global__ void resid_ln_kernel(float* __restrict__ Xio, const float* __restrict__ Yin,
                                const float* __restrict__ g, const float* __restrict__ b, int R) {
  int wid = threadIdx.x >> 5, lane = threadIdx.x & 31;
  int row = blockIdx.x * 4 + wid;
  if (row >= R) return;
  size_t o = (size_t)row * LATD + lane;
  float v  = Xio[o] + Yin[o];
  float s  = wave_sum32(v);
  float sq = wave_sum32(v * v);
  float mu  = s * (1.f / 32.f);
  float var = sq * (1.f / 32.f) - mu * mu;
  Xio[o] = (v - mu) * rsqrtf(var + EPSLN) * g[lane] + b[lane];
}

// ---------------- decoder output: [rows,64] @ [64,3] + b, one thread per row ----------------
__global__ void dec_out_kernel(const float* __restrict__ H, const float* __restrict__ W,
                               const float* __restrict__ Bv, float* __restrict__ Y, int rows) {
  int r = blockIdx.x * blockDim.x + threadIdx.x;
  if (r >= rows) return;
  const float* h = H + (size_t)r * HIDD;
  float a0 = Bv[0], a1 = Bv[1], a2 = Bv[2];
#pragma unroll
  for (int k = 0; k < HIDD; ++k) {
    float hv = h[k];
    a0 = fmaf(hv, W[k * 3 + 0], a0);
    a1 = fmaf(hv, W[k * 3 + 1], a1);
    a2 = fmaf(hv, W[k * 3 + 2], a2);
  }
  float* y = Y + (size_t)r * 3;
  y[0] = a0; y[1] = a1; y[2] = a2;
}

// =====================================================================
// kernel_launch
// Param leaf order assumption (insertion-order recursive flatten):
//   0 input_sequence  1 edge_index  2 edge_attr
//   3/4   enc_in.{W,b}
//   5..22 enc_layers[0..2]: eW,eb,nW,nb,ln_g,ln_b (6/layer)
//   23/24 enc_out.{W,b}
//   25..56 attn_layers[0..1]: qW,qb,kW,kb,vW,vb,oW,ob,
//          ff1W,ff1b,ff2W,ff2b,ln1g,ln1b,ln2g,ln2b (16/layer)
//   57..62 pred1.{W,b} pred2.{W,b} dec_in.{W,b}
//   63..80 dec_layers[0..2]   81/82 dec_out.{W,b}
// =====================================================================
extern "C" void kernel_launch(void* const* d_in, const int* in_sizes, int n_in,
                              void* d_out, int out_size, void* d_ws, size_t ws_size,
                              hipStream_t stream) {
  const float* in_seq = (const float*)d_in[0];
  const int*   eidx   = (const int*)d_in[1];
  const float* ea     = (const float*)d_in[2];
  const int E  = in_sizes[2];
  const int N  = in_sizes[0] / (BB * TT * 3);
  const int G  = BB * TT;       // 8 graphs in encoder
  const int GN = G * N;         // 160000
  const int BN = BB * N;        // 40000
  const int R  = BN * TT;       // 160000 transformer rows
  const int* src = eidx;
  const int* dst = eidx + E;

  auto P = [&](int i) { return (const float*)d_in[i]; };
  const int ENC_IN = 3, ENC_L = 5, ENC_OUT = 23, ATT_L = 25;
  const int PRED1 = 57, PRED2 = 59, DEC_IN = 61, DEC_L = 63, DEC_OUT = 81;

  // ---- workspace arena (floats) ----
  float* ws  = (float*)d_ws;
  float* deg = ws;                              // N
  float* A   = ws + 20480;                      // GN*HID  (10.24M floats)
  float* Bf  = A  + (size_t)GN * HIDD;          // GN*HID  — contiguous with A (ff1 temp)
  float* C   = Bf + (size_t)GN * HIDD;          // GN*LAT
  float* D   = C  + (size_t)GN * LATD;          // BN*T*LAT residual stream
  (void)ws_size; (void)n_in; (void)out_size;

  // ---- degrees (shared by encoder & decoder) ----
  deg_init_kernel<<<(N + 255) / 256, 256, 0, stream>>>(deg, N);
  deg_scatter_kernel<<<(E + 255) / 256, 256, 0, stream>>>(dst, deg, E);

  // ---- encoder ----
  enc_in_kernel<<<(GN * HIDD + 255) / 256, 256, 0, stream>>>(in_seq, P(ENC_IN), P(ENC_IN + 1), A, GN);

  const int ntiles = N / 16;  // 20000/16 = 1250
  for (int l = 0; l < 3; ++l) {
    const float *eW = P(ENC_L + 6 * l),     *eb = P(ENC_L + 6 * l + 1),
                *nW = P(ENC_L + 6 * l + 2), *nb = P(ENC_L + 6 * l + 3),
                *lg = P(ENC_L + 6 * l + 4), *lb = P(ENC_L + 6 * l + 5);
    agg_init_kernel<<<(GN * HIDD / 4 + 255) / 256, 256, 0, stream>>>(A, eW, eb, Bf, GN * HIDD / 4);
    edge_scatter_kernel<<<E, 64, 0, stream>>>(src, dst, ea, eW, eb, A, Bf, N, G);
    gnn_node_kernel<<<G * ntiles, 128, 0, stream>>>(Bf, deg, nW, nb, lg, lb, A, N, ntiles);
  }

  // z = x @ enc_out : [GN,64] -> [GN,32]
  wmma_linear_kernel<64, 32, 2, false><<<GN / 16, 64, 0, stream>>>(A, HIDD, P(ENC_OUT), P(ENC_OUT + 1), C, LATD, GN);
  // [B*T,N,LAT] -> [B,N,T,LAT]
  reorder_kernel<<<(BB * N * TT * (LATD / 4) + 255) / 256, 256, 0, stream>>>(C, D, N);

  // ---- transformer (2 layers, post-norm) ----
  float* qb = A;
  float* kb = A + (size_t)R * LATD;
  float* vb = Bf;
  float* ob = Bf + (size_t)R * LATD;
  for (int l = 0; l < 2; ++l) {
    int p = ATT_L + 16 * l;
    qkv_kernel<<<R / 16, 192, 0, stream>>>(D, P(p + 0), P(p + 1), P(p + 2), P(p + 3),
                                           P(p + 4), P(p + 5), qb, kb, vb);
    attn_kernel<<<(BN * NHEAD + 255) / 256, 256, 0, stream>>>(qb, kb, vb, ob, BN);
    wmma_linear_kernel<32, 32, 2, false><<<R / 16, 64, 0, stream>>>(ob, LATD, P(p + 6), P(p + 7), C, LATD, R);
    resid_ln_kernel<<<R / 4, 128, 0, stream>>>(D, C, P(p + 12), P(p + 13), R);
    // FFN: ff1 output [R,128] lives in the (dead) A||B region
    wmma_linear_kernel<32, 128, 4, true><<<R / 16, 128, 0, stream>>>(D, LATD, P(p + 8), P(p + 9), A, 128, R);
    wmma_linear_kernel<128, 32, 2, false><<<R / 16, 64, 0, stream>>>(A, 128, P(p + 10), P(p + 11), C, LATD, R);
    resid_ln_kernel<<<R / 4, 128, 0, stream>>>(D, C, P(p + 14), P(p + 15), R);
  }

  // ---- prediction head: z_last = D[:, T-1, :] (strided read) ----
  wmma_linear_kernel<32, 64, 4, true ><<<BN / 16, 128, 0, stream>>>(D + (TT - 1) * LATD, TT * LATD, P(PRED1), P(PRED1 + 1), A, HIDD, BN);
  wmma_linear_kernel<64, 32, 2, false><<<BN / 16, 64, 0, stream>>>(A, HIDD, P(PRED2), P(PRED2 + 1), Bf, LATD, BN);
  wmma_linear_kernel<32, 64, 4, false><<<BN / 16, 128, 0, stream>>>(Bf, LATD, P(DEC_IN), P(DEC_IN + 1), A, HIDD, BN);

  // ---- decoder GNN (G = 2 graphs, x in A, agg in Bf) ----
  for (int l = 0; l < 3; ++l) {
    const float *eW = P(DEC_L + 6 * l),     *eb = P(DEC_L + 6 * l + 1),
                *nW = P(DEC_L + 6 * l + 2), *nb = P(DEC_L + 6 * l + 3),
                *lg = P(DEC_L + 6 * l + 4), *lb = P(DEC_L + 6 * l + 5);
    agg_init_kernel<<<(BN * HIDD / 4 + 255) / 256, 256, 0, stream>>>(A, eW, eb, Bf, BN * HIDD / 4);
    edge_scatter_kernel<<<E, 64, 0, stream>>>(src, dst, ea, eW, eb, A, Bf, N, BB);
    gnn_node_kernel<<<BB * ntiles, 128, 0, stream>>>(Bf, deg, nW, nb, lg, lb, A, N, ntiles);
  }

  dec_out_kernel<<<(BN + 255) / 256, 256, 0, stream>>>(A, P(DEC_OUT), P(DEC_OUT + 1), (float*)d_out, BN);
}